// MSTAGNN_64218351010249
// MI455X (gfx1250) — compile-verified
//
#include <hip/hip_runtime.h>
#include <hip/hip_bf16.h>
#include <cstdint>

#define HID 64
#define HEADS 8
#define KHOPS 2

typedef __attribute__((ext_vector_type(2))) float v2f;
typedef __attribute__((ext_vector_type(8))) float v8f;

__device__ __forceinline__ void atomic_add_f32(float* p, float v) {
  // lowers to global_atomic_add_f32 (agent scope, relaxed) — resolves in L2
  __hip_atomic_fetch_add(p, v, __ATOMIC_RELAXED, __HIP_MEMORY_SCOPE_AGENT);
}

// ---------------- utility ----------------
__global__ void zero_f32(float* __restrict__ p, size_t n) {
  size_t i = (size_t)blockIdx.x * blockDim.x + threadIdx.x;
  if (i < n) p[i] = 0.0f;
}

// deg[col[e]] += 1  (in-degree histogram)
__global__ void deg_scatter(const long long* __restrict__ ei, float* __restrict__ deg, int E) {
  int e = blockIdx.x * blockDim.x + threadIdx.x;
  if (e < E) atomic_add_f32(&deg[(int)ei[(size_t)E + e]], 1.0f);
}

// deg -> deg^-0.5 (0 if deg==0), in place
__global__ void deg_finalize(float* __restrict__ deg, int N) {
  int i = blockIdx.x * blockDim.x + threadIdx.x;
  if (i < N) {
    float d = deg[i];
    deg[i] = d > 0.0f ? (1.0f / sqrtf(d)) : 0.0f;
  }
}

// ---------------- fused QKV projection via V_WMMA_F32_16X16X4_F32 ----------------
// One wave computes a 16-row tile of Q, K, V (all 64 cols each).
// A operand (16x4 f32): lane<16 holds (m=lane, k, k+1); lane>=16 holds (m=lane-16, k+2, k+3).
// B operand (4x16 f32): same K split per lane half, lane&15 selects column n.
// C/D (16x16 f32): VGPR i -> row (i + 8*(lane>>4)), col (lane&15).
__global__ __launch_bounds__(256) void qkv_wmma(
    const float* __restrict__ x,
    const float* __restrict__ Wq, const float* __restrict__ bq,
    const float* __restrict__ Wk, const float* __restrict__ bk,
    const float* __restrict__ Wv, const float* __restrict__ bv,
    const float* __restrict__ hopwise,
    float* __restrict__ Q, float* __restrict__ K, float* __restrict__ V,
    float* __restrict__ hidden, int N)
{
  const int lane = threadIdx.x & 31;
  const int wib  = threadIdx.x >> 5;
  const int rowBase = (blockIdx.x * 8 + wib) * 16;
  if (rowBase >= N) return;                 // wave-uniform exit: EXEC stays all-1s
  const int nsub = lane & 15;
  const int half = lane >> 4;

  const int rowA  = rowBase + nsub;
  const int rowAc = rowA < N ? rowA : (N - 1);   // clamp loads; stores are guarded

  // Preload the full 16x64 A tile for this wave: 16 K-steps of (k, k+1) per lane
  v2f areg[16];
  const float* xr = x + (size_t)rowAc * HID;
  #pragma unroll
  for (int t = 0; t < 16; ++t) {
    const int k = t * 4 + half * 2;
    areg[t][0] = xr[k];
    areg[t][1] = xr[k + 1];
  }

  const float h0 = hopwise[0];

  const float* Ws[3] = {Wq, Wk, Wv};
  const float* bs[3] = {bq, bk, bv};
  float*       Os[3] = {Q, K, V};

  #pragma unroll
  for (int w = 0; w < 3; ++w) {
    const float* __restrict__ W = Ws[w];
    const float* __restrict__ b = bs[w];
    float* __restrict__ O = Os[w];
    #pragma unroll
    for (int nt = 0; nt < 4; ++nt) {
      v8f acc;
      #pragma unroll
      for (int i = 0; i < 8; ++i) acc[i] = 0.0f;
      #pragma unroll
      for (int t = 0; t < 16; ++t) {
        const int k = t * 4 + half * 2;
        v2f bb;
        bb[0] = W[(size_t)k * HID + nt * 16 + nsub];
        bb[1] = W[(size_t)(k + 1) * HID + nt * 16 + nsub];
        acc = __builtin_amdgcn_wmma_f32_16x16x4_f32(
            false, areg[t], false, bb, (short)0, acc, false, false);
      }
      const int   col  = nt * 16 + nsub;
      const float bias = b[col];
      #pragma unroll
      for (int i = 0; i < 8; ++i) {
        const int r = rowBase + i + half * 8;
        if (r < N) {
          float v0 = acc[i] + bias;
          if (w < 2) v0 = (v0 > 0.0f) ? (v0 + 1.0f) : expf(v0);  // 1 + elu
          O[(size_t)r * HID + col] = v0;
          if (w == 2) hidden[(size_t)r * HID + col] = v0 * h0;   // hopwise[0] * V
        }
      }
    }
  }
}

// ---------------- M0[n,h,i,j] = K[n,h,i] * V[n,h,j] ----------------
__global__ void m0_init(const float* __restrict__ K, const float* __restrict__ V,
                        float* __restrict__ M, size_t total) {
  size_t t = (size_t)blockIdx.x * blockDim.x + threadIdx.x;
  if (t >= total) return;
  size_t n = t >> 9;
  int r = (int)(t & 511);
  int h = r >> 6, i = (r >> 3) & 7, j = r & 7;
  M[t] = K[n * HID + h * 8 + i] * V[n * HID + h * 8 + j];
}

// ---------------- one hop of scatter propagation (wave per edge) ----------------
// Mout[col] += norm * Min[row] (512 f32), Kout[col] += norm * Kin[row] (64 f32)
__global__ __launch_bounds__(256) void propagate(
    const long long* __restrict__ ei, const float* __restrict__ dinv,
    const float* __restrict__ Min, const float* __restrict__ Kin,
    float* __restrict__ Mout, float* __restrict__ Kout, int E)
{
  const int gw   = (int)(((size_t)blockIdx.x * blockDim.x + threadIdx.x) >> 5);
  const int lane = threadIdx.x & 31;
  if (gw >= E) return;                       // wave-uniform
  const int row = (int)ei[gw];
  const int col = (int)ei[(size_t)E + gw];
  const float norm = dinv[row] * dinv[col];
  const float* __restrict__ ms = Min + (size_t)row * 512;
  float* __restrict__       md = Mout + (size_t)col * 512;
  #pragma unroll
  for (int t = lane; t < 512; t += 32) atomic_add_f32(&md[t], norm * ms[t]);
  const float* __restrict__ ks = Kin + (size_t)row * HID;
  float* __restrict__       kd = Kout + (size_t)col * HID;
  #pragma unroll
  for (int t = lane; t < HID; t += 32) atomic_add_f32(&kd[t], norm * ks[t]);
}

// ---------------- combine: hidden += gamma * (Q.M) / (Q.K + eps) ----------------
__global__ void combine(const float* __restrict__ Q, const float* __restrict__ Kp,
                        const float* __restrict__ M,
                        const float* __restrict__ hopwise, const float* __restrict__ headwise,
                        float* __restrict__ hidden, int N, int hop)
{
  const int t = blockIdx.x * blockDim.x + threadIdx.x;
  if (t >= N * HEADS) return;
  const int n = t >> 3, h = t & 7;

  // layerwise = softmax over heads of headwise[:, hop]
  float vals[HEADS], mx = -3.0e38f;
  #pragma unroll
  for (int hh = 0; hh < HEADS; ++hh) {
    vals[hh] = headwise[hh * KHOPS + hop];
    mx = fmaxf(mx, vals[hh]);
  }
  float s = 0.0f;
  #pragma unroll
  for (int hh = 0; hh < HEADS; ++hh) s += expf(vals[hh] - mx);
  const float gamma = hopwise[hop + 1] * expf(vals[h] - mx) / s;

  const float* __restrict__ q = Q  + (size_t)n * HID + h * 8;
  const float* __restrict__ k = Kp + (size_t)n * HID + h * 8;
  float qv[8];
  float C = 1e-5f;
  #pragma unroll
  for (int i = 0; i < 8; ++i) { qv[i] = q[i]; C += qv[i] * k[i]; }
  const float invC = 1.0f / C;

  const float* __restrict__ Mn = M + ((size_t)n * HEADS + h) * 64;
  float* __restrict__ hd = hidden + (size_t)n * HID + h * 8;
  #pragma unroll
  for (int j = 0; j < 8; ++j) {
    float Hacc = 0.0f;
    #pragma unroll
    for (int i = 0; i < 8; ++i) Hacc += qv[i] * Mn[i * 8 + j];
    hd[j] += gamma * Hacc * invC;
  }
}

// ---------------- output projection via WMMA: out = hidden @ Wo + bo ----------------
__global__ __launch_bounds__(256) void outproj_wmma(
    const float* __restrict__ hidden, const float* __restrict__ Wo,
    const float* __restrict__ bo, float* __restrict__ out, int N)
{
  const int lane = threadIdx.x & 31;
  const int wib  = threadIdx.x >> 5;
  const int rowBase = (blockIdx.x * 8 + wib) * 16;
  if (rowBase >= N) return;
  const int nsub = lane & 15;
  const int half = lane >> 4;
  const int rowA  = rowBase + nsub;
  const int rowAc = rowA < N ? rowA : (N - 1);

  v2f areg[16];
  const float* hr = hidden + (size_t)rowAc * HID;
  #pragma unroll
  for (int t = 0; t < 16; ++t) {
    const int k = t * 4 + half * 2;
    areg[t][0] = hr[k];
    areg[t][1] = hr[k + 1];
  }

  #pragma unroll
  for (int nt = 0; nt < 4; ++nt) {
    v8f acc;
    #pragma unroll
    for (int i = 0; i < 8; ++i) acc[i] = 0.0f;
    #pragma unroll
    for (int t = 0; t < 16; ++t) {
      const int k = t * 4 + half * 2;
      v2f bb;
      bb[0] = Wo[(size_t)k * HID + nt * 16 + nsub];
      bb[1] = Wo[(size_t)(k + 1) * HID + nt * 16 + nsub];
      acc = __builtin_amdgcn_wmma_f32_16x16x4_f32(
          false, areg[t], false, bb, (short)0, acc, false, false);
    }
    const int   col  = nt * 16 + nsub;
    const float bias = bo[col];
    #pragma unroll
    for (int i = 0; i < 8; ++i) {
      const int r = rowBase + i + half * 8;
      if (r < N) out[(size_t)r * HID + col] = acc[i] + bias;
    }
  }
}

// ---------------- host-side orchestration ----------------
extern "C" void kernel_launch(void* const* d_in, const int* in_sizes, int n_in,
                              void* d_out, int out_size, void* d_ws, size_t ws_size,
                              hipStream_t stream) {
  const float*     x        = (const float*)d_in[0];
  const long long* ei       = (const long long*)d_in[1];   // int64 edge_index [2,E]
  const float*     Wq       = (const float*)d_in[3];
  const float*     bq       = (const float*)d_in[4];
  const float*     Wk       = (const float*)d_in[5];
  const float*     bk       = (const float*)d_in[6];
  const float*     Wv       = (const float*)d_in[7];
  const float*     bv       = (const float*)d_in[8];
  const float*     Wo       = (const float*)d_in[9];
  const float*     bo       = (const float*)d_in[10];
  const float*     hopwise  = (const float*)d_in[11];
  const float*     headwise = (const float*)d_in[12];
  float*           out      = (float*)d_out;

  const int N = in_sizes[0] / HID;
  const int E = in_sizes[1] / 2;

  // workspace carve (everything fits comfortably in MI455X's 192MB L2)
  char* ws = (char*)d_ws;
  size_t off = 0;
  auto carve = [&](size_t bytes) -> float* {
    float* p = (float*)(ws + off);
    off += (bytes + 255) & ~(size_t)255;
    return p;
  };
  float* Q    = carve((size_t)N * HID * 4);
  float* K    = carve((size_t)N * HID * 4);
  float* V    = carve((size_t)N * HID * 4);
  float* hid  = carve((size_t)N * HID * 4);
  float* KpA  = carve((size_t)N * HID * 4);
  float* KpB  = carve((size_t)N * HID * 4);
  float* dinv = carve((size_t)N * 4);
  float* MA   = carve((size_t)N * 512 * 4);
  float* MB   = carve((size_t)N * 512 * 4);
  (void)ws_size; (void)n_in; (void)out_size;

  const int T = 256;
  auto nb = [](size_t n, int t) { return (unsigned)((n + (size_t)t - 1) / (size_t)t); };

  // degrees -> deg^-1/2
  zero_f32<<<nb((size_t)N, T), T, 0, stream>>>(dinv, (size_t)N);
  deg_scatter<<<nb((size_t)E, T), T, 0, stream>>>(ei, dinv, E);
  deg_finalize<<<nb((size_t)N, T), T, 0, stream>>>(dinv, N);

  // fused QKV projection (WMMA) + hidden init
  qkv_wmma<<<nb((size_t)N, 128), 256, 0, stream>>>(x, Wq, bq, Wk, bk, Wv, bv,
                                                   hopwise, Q, K, V, hid, N);

  const size_t mtot = (size_t)N * 512;
  const size_t ktot = (size_t)N * HID;
  m0_init<<<nb(mtot, T), T, 0, stream>>>(K, V, MA, mtot);

  // hop 0: MA,K -> MB,KpB
  zero_f32<<<nb(mtot, T), T, 0, stream>>>(MB, mtot);
  zero_f32<<<nb(ktot, T), T, 0, stream>>>(KpB, ktot);
  propagate<<<nb((size_t)E * 32, T), T, 0, stream>>>(ei, dinv, MA, K, MB, KpB, E);
  combine<<<nb((size_t)N * HEADS, T), T, 0, stream>>>(Q, KpB, MB, hopwise, headwise, hid, N, 0);

  // hop 1: MB,KpB -> MA,KpA
  zero_f32<<<nb(mtot, T), T, 0, stream>>>(MA, mtot);
  zero_f32<<<nb(ktot, T), T, 0, stream>>>(KpA, ktot);
  propagate<<<nb((size_t)E * 32, T), T, 0, stream>>>(ei, dinv, MB, KpB, MA, KpA, E);
  combine<<<nb((size_t)N * HEADS, T), T, 0, stream>>>(Q, KpA, MA, hopwise, headwise, hid, N, 1);

  // output projection (WMMA)
  outproj_wmma<<<nb((size_t)N, 128), 256, 0, stream>>>(hid, Wo, bo, out, N);
}